// LinODEnet_75668733821391
// MI455X (gfx1250) — compile-verified
//
#include <hip/hip_runtime.h>

// ---------------------------------------------------------------------------
// LinODEnet on MI455X (gfx1250): single persistent workgroup, f32 WMMA.
//
// expm(dt*K) @ z is evaluated as the action of the exponential on the batched
// state (Taylor/Horner, ||dt*K|| <= ~0.02 so 8 terms << fp32 eps), turning
// 1536 dense 160^3 Pade matmuls (~100 GFLOP) into shared-weight
// (32 x 160) @ (160 x 160) WMMA GEMMs (~1.6 GFLOP total) with per-row dt^n/n!
// coefficients. The workload is latency-bound: 48-step serial chain, 20 KB
// state. So: one workgroup (10 waves, wave32), activations LDS-resident,
// V_WMMA_F32_16X16X4_F32 for every GEMM, weights streamed from the 192 MB L2,
// and the constant ODE kernel staged once into LDS with CDNA5
// global_load_async_to_lds_b128 (ASYNCcnt) in a bank-conflict-free padded
// layout (row stride 164 floats -> 36 banks -> 16 fragment rows on 16
// distinct banks).
//
// Static tile map (blockDim=320, tilesM=2): wave w owns (tm=w&1, tn=w>>1) and
// (tm, tn+5). Both tiles share the A fragment -> one A load feeds two WMMAs.
// ---------------------------------------------------------------------------

typedef __attribute__((ext_vector_type(2))) float v2f;
typedef __attribute__((ext_vector_type(8))) float v8f;

#define EP_BIAS    1
#define EP_RELU    2
#define EP_SIGMOID 4
#define EP_INITC   8   // accumulate onto existing C (residual / GRU gate pairs)

#define LD160 164      // padded row stride for 160-wide buffers (floats)
#define LD128 132      // padded row stride for 128-wide buffers (floats)

struct LinodeParams {
  const float* T;   // (32,48)
  const float* X;   // (32,48,96) with NaNs
  const float* K;   // (160,160)
  const float* z0;  // (160)
  const float* emb; // (32)
  const float* wih; // (384,128)
  const float* whh; // (384,128)
  const float* bih; // (384)
  const float* bhh; // (384)
  const float* encW[8];  // W1_0,b1_0,W2_0,b2_0,W1_1,b1_1,W2_1,b2_1
  const float* decW[8];
  float* out;       // (32,48,96)
};

__device__ __forceinline__ float sigmoidf_(float v) { return 1.0f / (1.0f + __expf(-v)); }

__device__ __forceinline__ v8f wmma4(v2f a, v2f b, v8f c) {
  return __builtin_amdgcn_wmma_f32_16x16x4_f32(false, a, false, b, (short)0, c,
                                               false, false);
}

__device__ __forceinline__ void epilogue_tile(float* __restrict__ C, v8f c,
                                              int mbase, int n, int ldc,
                                              const float* bias, int flags) {
  const float bv = (flags & EP_BIAS) ? bias[n] : 0.f;
  #pragma unroll
  for (int r = 0; r < 8; ++r) {
    float v = c[r] + bv;
    if (flags & EP_INITC)   v += C[(mbase + r) * ldc + n];
    if (flags & EP_RELU)    v = fmaxf(v, 0.f);
    if (flags & EP_SIGMOID) v = sigmoidf_(v);
    C[(mbase + r) * ldc + n] = v;
  }
}

// C[32xN] = epilogue( A[32xK] * B^T ),  B[k][n] = Bw[n*ldb + k]
// (reference matmuls are x @ W.T, W stored (N,K) row-major).
// In-place (C==A) safe: full K-loop reads, workgroup barrier, then writeback.
__device__ __forceinline__ void gemm_ws(float* __restrict__ C,
                                        const float* __restrict__ A,
                                        const float* __restrict__ Bw,
                                        int tilesN, int K, int ldc, int lda, int ldb,
                                        const float* bias, int flags)
{
  const int tid  = threadIdx.x;
  const int lane = tid & 31;
  const int wave = __builtin_amdgcn_readfirstlane(tid >> 5);  // scalar -> s_cbranch
  const int lm   = lane & 15;     // A row / B,C column within tile
  const int half = lane >> 4;     // 0: K{0,1}, 1: K{2,3} striping
  const int tm   = wave & 1;      // M tile (M=32 -> tilesM=2)
  const int tnA  = wave >> 1;     // first N tile (0..4, always valid)
  const int tnB  = tnA + 5;       // second N tile, shares A fragment
  const bool hasB = (tnB < tilesN);

  const float* Arow = A  + (tm  * 16 + lm) * lda + (half << 1);
  const float* B0   = Bw + (tnA * 16 + lm) * ldb + (half << 1);
  const float* B1   = Bw + (tnB * 16 + lm) * ldb + (half << 1);

  v8f c0 = {0.f, 0.f, 0.f, 0.f, 0.f, 0.f, 0.f, 0.f};
  v8f c1 = {0.f, 0.f, 0.f, 0.f, 0.f, 0.f, 0.f, 0.f};

  if (hasB) {
    #pragma unroll 4
    for (int k0 = 0; k0 < K; k0 += 4) {
      v2f a, b, b2;
      a.x  = Arow[k0]; a.y  = Arow[k0 + 1];
      b.x  = B0[k0];   b.y  = B0[k0 + 1];
      b2.x = B1[k0];   b2.y = B1[k0 + 1];
      c0 = wmma4(a, b,  c0);      // one A fragment feeds two WMMAs
      c1 = wmma4(a, b2, c1);
    }
  } else {
    #pragma unroll 4
    for (int k0 = 0; k0 < K; k0 += 4) {
      v2f a, b;
      a.x = Arow[k0]; a.y = Arow[k0 + 1];
      b.x = B0[k0];   b.y = B0[k0 + 1];
      c0 = wmma4(a, b, c0);
    }
  }
  __syncthreads();   // all A reads complete before any C writes (in-place safety)
  epilogue_tile(C, c0, tm * 16 + (half << 3), tnA * 16 + lm, ldc, bias, flags);
  if (hasB)
    epilogue_tile(C, c1, tm * 16 + (half << 3), tnB * 16 + lm, ldc, bias, flags);
  __syncthreads();
}

// x = x + relu(x @ W1.T + b1) @ W2.T + b2     (L = 160)
__device__ __forceinline__ void resnet_block(float* Xb, float* Hb,
                                             const float* W1, const float* b1,
                                             const float* W2, const float* b2)
{
  gemm_ws(Hb, Xb, W1, 10, 160, LD160, LD160, 160, b1, EP_BIAS | EP_RELU);
  gemm_ws(Xb, Hb, W2, 10, 160, LD160, LD160, 160, b2, EP_BIAS | EP_INITC);
}

extern "C" __global__ __launch_bounds__(320)
void linodenet_gfx1250_kernel(LinodeParams p)
{
  extern __shared__ float smem[];
  // LDS arena (floats), all GEMM operands bank-conflict-free padded.
  float* Z    = smem;              // 32 x LD160  persistent latent state
  float* A0   = smem + 5248;       // 32 x LD160  z_pre -> resnet act -> enc input
  float* A1   = smem + 10496;      // 32 x LD160  Taylor term / resnet hidden
  float* XPRE = smem + 15744;      // 32 x LD128
  float* Y    = smem + 19968;      // 32 x LD128
  float* INg  = smem + 24192;      // 32 x LD128
  float* HNg  = smem + 28416;      // 32 x LD128
  float* XPO  = smem + 32640;      // 32 x LD128
  float* dts  = smem + 36864;      // 32
  float* dtp  = smem + 36896;      // 32
  float* KLDS = smem + 36928;      // 160 x LD160 staged ODE kernel
  float* Rg   = A1;                // GRU-phase aliases (ld = LD128)
  float* ZGg  = A0;

  const int tid  = threadIdx.x;
  const int nthr = blockDim.x;

  // ---- stage constant ODE kernel K into LDS via CDNA5 async copies ----
  // 160 rows x 40 b128 chunks, repacked to the padded LD160 layout.
  {
    const uint32_t kbase = (uint32_t)(uintptr_t)KLDS;   // LDS byte offset (addr[31:0])
    for (int c = tid; c < 6400; c += nthr) {
      const int r = c / 40, q = c - r * 40;
      const float* src = p.K + r * 160 + q * 4;
      const uint32_t dst = kbase + (uint32_t)(r * LD160 + q * 4) * 4u;
      asm volatile("global_load_async_to_lds_b128 %0, %1, off"
                   :: "v"(dst), "v"(src) : "memory");
    }
    asm volatile("s_wait_asynccnt 0x0" ::: "memory");
  }
  // Z[b,:] = z0
  for (int i = tid; i < 32 * 160; i += nthr) {
    const int b = i / 160, j = i - b * 160;
    Z[b * LD160 + j] = p.z0[j];
  }
  __syncthreads();

  #pragma unroll 1
  for (int t = 0; t < 48; ++t) {
    // dt_b = T[b,t] - T[b,t-1]   (0 at t=0: expm(0)=I)
    if (tid < 32) {
      const float dt = (t == 0) ? 0.f : (p.T[tid * 48 + t] - p.T[tid * 48 + t - 1]);
      dts[tid] = dt; dtp[tid] = 1.f;
    }
    __syncthreads();

    // ---- z_pre = expm(dt_b*K) z_post via batched Taylor action ----
    for (int i = tid; i < 5120; i += nthr) {
      const int b = i / 160, j = i - b * 160;
      const float z = Z[b * LD160 + j];
      A0[b * LD160 + j] = z; A1[b * LD160 + j] = z;
    }
    __syncthreads();
    #pragma unroll 1
    for (int n = 1; n <= 8; ++n) {
      // v <- v @ K^T (B[k][n] = K[n,k]), B served from LDS-staged KLDS
      gemm_ws(A1, A1, KLDS, 10, 160, LD160, LD160, LD160, nullptr, 0);
      if (tid < 32) dtp[tid] *= dts[tid] * (1.f / (float)n);
      __syncthreads();
      for (int i = tid; i < 5120; i += nthr) {
        const int b = i / 160, j = i - b * 160;
        A0[b * LD160 + j] += dtp[b] * A1[b * LD160 + j];
      }
      __syncthreads();
    }

    // ---- decoder resnet (2 blocks); x_pre = out[:, :128] ----
    resnet_block(A0, A1, p.decW[0], p.decW[1], p.decW[2], p.decW[3]);
    resnet_block(A0, A1, p.decW[4], p.decW[5], p.decW[6], p.decW[7]);
    for (int i = tid; i < 4096; i += nthr) {
      const int b = i >> 7, j = i & 127;
      XPRE[b * LD128 + j] = A0[b * LD160 + j];
    }
    __syncthreads();

    // ---- impute: y = isnan(x_obs) ? x_pre : x_obs (cols >= 96 are NaN pad) ----
    for (int i = tid; i < 4096; i += nthr) {
      const int b = i >> 7, j = i & 127;
      float yv = XPRE[b * LD128 + j];
      if (j < 96) {
        const float xo = p.X[(b * 48 + t) * 96 + j];
        if (!(xo != xo)) yv = xo;
      }
      Y[b * LD128 + j] = yv;
    }
    __syncthreads();

    // ---- GRU cell (hidden = x_pre): 32x128x128 WMMA GEMM pairs ----
    gemm_ws(Rg,  Y,    p.wih,             8,128, LD128,LD128,128, p.bih,       EP_BIAS);
    gemm_ws(Rg,  XPRE, p.whh,             8,128, LD128,LD128,128, p.bhh,       EP_BIAS|EP_INITC|EP_SIGMOID);
    gemm_ws(ZGg, Y,    p.wih + 128*128,   8,128, LD128,LD128,128, p.bih + 128, EP_BIAS);
    gemm_ws(ZGg, XPRE, p.whh + 128*128,   8,128, LD128,LD128,128, p.bhh + 128, EP_BIAS|EP_INITC|EP_SIGMOID);
    gemm_ws(INg, Y,    p.wih + 2*128*128, 8,128, LD128,LD128,128, p.bih + 256, EP_BIAS);
    gemm_ws(HNg, XPRE, p.whh + 2*128*128, 8,128, LD128,LD128,128, p.bhh + 256, EP_BIAS);
    for (int i = tid; i < 4096; i += nthr) {
      const int idx = (i >> 7) * LD128 + (i & 127);
      const float ng = tanhf(INg[idx] + Rg[idx] * HNg[idx]);
      const float zg = ZGg[idx];
      XPO[idx] = (1.f - zg) * ng + zg * XPRE[idx];
    }
    __syncthreads();

    // ---- emit x_post[:, :96] -> out[b, t, :] ----
    for (int i = tid; i < 32 * 96; i += nthr) {
      const int b = i / 96, j = i - b * 96;
      p.out[(b * 48 + t) * 96 + j] = XPO[b * LD128 + j];
    }
    __syncthreads();

    // ---- encoder resnet on [x_post, emb_pad] -> new z_post ----
    for (int i = tid; i < 5120; i += nthr) {
      const int b = i / 160, j = i - b * 160;
      A0[b * LD160 + j] = (j < 128) ? XPO[b * LD128 + j] : p.emb[j - 128];
    }
    __syncthreads();
    resnet_block(A0, A1, p.encW[0], p.encW[1], p.encW[2], p.encW[3]);
    resnet_block(A0, A1, p.encW[4], p.encW[5], p.encW[6], p.encW[7]);
    for (int i = tid; i < 5120; i += nthr) {
      const int b = i / 160, j = i - b * 160;
      Z[b * LD160 + j] = A0[b * LD160 + j];
    }
    __syncthreads();
  }
}

extern "C" void kernel_launch(void* const* d_in, const int* in_sizes, int n_in,
                              void* d_out, int out_size, void* d_ws, size_t ws_size,
                              hipStream_t stream) {
  (void)in_sizes; (void)n_in; (void)out_size; (void)d_ws; (void)ws_size;
  LinodeParams p;
  p.T   = (const float*)d_in[0];
  p.X   = (const float*)d_in[1];
  p.K   = (const float*)d_in[2];
  p.z0  = (const float*)d_in[3];
  p.emb = (const float*)d_in[4];
  p.wih = (const float*)d_in[5];
  p.whh = (const float*)d_in[6];
  p.bih = (const float*)d_in[7];
  p.bhh = (const float*)d_in[8];
  // pytree-leaf order: enc_params then dec_params, each (W1,b1,W2,b2) x NBLOCKS
  for (int i = 0; i < 8; ++i) p.encW[i] = (const float*)d_in[9 + i];
  for (int i = 0; i < 8; ++i) p.decW[i] = (const float*)d_in[17 + i];
  p.out = (float*)d_out;

  const size_t shmem = 63168 * sizeof(float);   // ~247 KB of the 320 KB WGP LDS
  hipFuncSetAttribute((const void*)linodenet_gfx1250_kernel,
                      hipFuncAttributeMaxDynamicSharedMemorySize, (int)shmem);
  linodenet_gfx1250_kernel<<<dim3(1), dim3(320), shmem, stream>>>(p);
}